// SparseMoE_63591285784864
// MI455X (gfx1250) — compile-verified
//
#include <hip/hip_runtime.h>
#include <stdint.h>

// ---------------- problem constants (from reference) ----------------
#define EMBED   1024
#define NEXP    8
#define HIDDEN  4096
#define NTOK    4096           // B*T = 4*1024

// ---------------- GEMM tiling ----------------
#define KC   16                // K-slab per LDS stage (double buffered)
#define LDA  20                // KC + 4 pad  -> conflict-free A-fragment reads
#define LDB  136               // 128 + 8 pad -> conflict-free B-fragment reads
#define ABUF (128 * LDA)       // floats per A buffer
#define BBUF (KC * LDB)        // floats per B buffer

typedef float v2f __attribute__((ext_vector_type(2)));
typedef float v4f __attribute__((ext_vector_type(4)));
typedef float v8f __attribute__((ext_vector_type(8)));

#ifndef __has_builtin
#define __has_builtin(x) 0
#endif

// ---- CDNA5 async global->LDS path (ASYNCcnt), guarded so compile never breaks
#if __has_builtin(__builtin_amdgcn_global_load_async_to_lds_b128)
#define HAVE_ASYNC_LDS 1
// exact parameter types per hipcc diagnostic:
//   arg0: int __attribute__((vector_size(16))) addrspace(1)*   (global src)
//   arg1: int __attribute__((vector_size(16))) addrspace(3)*   (LDS dst)
typedef int v4i_gcc __attribute__((vector_size(16)));
typedef __attribute__((address_space(1))) v4i_gcc gv4i;
typedef __attribute__((address_space(3))) v4i_gcc lv4i;
#else
#define HAVE_ASYNC_LDS 0
#endif

#if HAVE_ASYNC_LDS
# if __has_builtin(__builtin_amdgcn_s_wait_asynccnt)
#  define WAIT_ASYNC(n) __builtin_amdgcn_s_wait_asynccnt(n)
# else
#  define WAIT_ASYNC(n) asm volatile("s_wait_asynccnt " #n ::: "memory")
# endif
#else
# define WAIT_ASYNC(n) ((void)0)
#endif

__device__ __forceinline__ void ld16(const float* gp, float* lp)
{
#if HAVE_ASYNC_LDS
    // ASYNCcnt-tracked DMA: global -> LDS, no VGPR round trip
    __builtin_amdgcn_global_load_async_to_lds_b128(
        (gv4i*)(uintptr_t)gp, (lv4i*)lp, 0, 0);
#else
    *(v4f*)lp = *(const v4f*)gp;
#endif
}

// stage one K-chunk: A tile 128 x KC (row stride lda), B tile KC x 128 (row stride ldb)
// 4 x b128 per thread  =>  4 async ops per wave per chunk
__device__ __forceinline__ void stage_chunk(const float* __restrict__ Ag, int lda,
                                            const float* __restrict__ Bg, int ldb,
                                            int k0, float* As, float* Bs, int tid)
{
#pragma unroll
    for (int i = 0; i < 2; ++i) {
        const int idx = tid + i * 256;            // float4 id, 0..511
        const int r = idx >> 2, c = (idx & 3) << 2;
        ld16(&Ag[(size_t)r * lda + k0 + c], &As[r * LDA + c]);
    }
#pragma unroll
    for (int i = 0; i < 2; ++i) {
        const int idx = tid + i * 256;
        const int r = idx >> 5, c = (idx & 31) << 2;
        ld16(&Bg[(size_t)(k0 + r) * ldb + c], &Bs[r * LDB + c]);
        if (i == 0)  // prefetch the weight stream two chunks ahead (speculative)
            __builtin_prefetch(&Bg[(size_t)(k0 + 2 * KC + r) * ldb + c], 0, 1);
    }
}

// one KC-slab of WMMAs: 4 K-steps x 8 M-tiles, fp32 16x16x4
__device__ __forceinline__ void compute_chunk(const float* As, const float* Bs,
                                              v8f acc[8], int l15, int hh, int jw)
{
#pragma unroll
    for (int kk = 0; kk < KC; kk += 4) {
        const int kb = kk + 2 * hh;               // lanes>=16 take K+2,K+3
        v2f bf; bf.x = Bs[kb * LDB + jw]; bf.y = Bs[(kb + 1) * LDB + jw];
#pragma unroll
        for (int mt = 0; mt < 8; ++mt) {
            const float* ap = &As[(mt * 16 + l15) * LDA + kb];
            v2f af; af.x = ap[0]; af.y = ap[1];
            acc[mt] = __builtin_amdgcn_wmma_f32_16x16x4_f32(
                false, af, false, bf, (short)0, acc[mt], false, false);
        }
    }
}

// double-buffered pipeline: stage chunk c+1 while computing chunk c
__device__ __forceinline__ void gemm_pipeline(const float* Ag, int lda,
                                              const float* Bg, int ldb, int K,
                                              float* As, float* Bs, v8f acc[8],
                                              int tid, int l15, int hh, int jw)
{
    const int NCH = K / KC;
    stage_chunk(Ag, lda, Bg, ldb, 0, As, Bs, tid);
    for (int c = 0; c < NCH; ++c) {
        const float* Ac = As + (c & 1) * ABUF;
        const float* Bc = Bs + (c & 1) * BBUF;
        if (c + 1 < NCH) {
            stage_chunk(Ag, lda, Bg, ldb, (c + 1) * KC,
                        As + ((c + 1) & 1) * ABUF, Bs + ((c + 1) & 1) * BBUF, tid);
            WAIT_ASYNC(4);    // chunk c complete; chunk c+1 may stay in flight
        } else {
            WAIT_ASYNC(0);
        }
        __syncthreads();      // all waves see chunk c in LDS
        compute_chunk(Ac, Bc, acc, l15, hh, jw);
        __syncthreads();      // done reading before buffer is overwritten
    }
}

// =====================================================================
// Router: scores = x@Wr + br ; top-2 ; mask-softmax over all 8 experts.
// One wave (32 lanes) per token; 8 tokens per 256-thread block.
// =====================================================================
__global__ __launch_bounds__(256) void moe_router(
    const float* __restrict__ x, const float* __restrict__ Wr,
    const float* __restrict__ br, float* __restrict__ wts)
{
    const int lane = threadIdx.x & 31;
    const int wave = threadIdx.x >> 5;
    const int t = blockIdx.x * 8 + wave;

    float acc[NEXP];
#pragma unroll
    for (int e = 0; e < NEXP; ++e) acc[e] = 0.f;

    const float* xr = x + (size_t)t * EMBED;
    for (int k = lane; k < EMBED; k += 32) {
        const float xv = xr[k];
#pragma unroll
        for (int e = 0; e < NEXP; ++e)
            acc[e] = fmaf(xv, Wr[k * NEXP + e], acc[e]);
    }
#pragma unroll
    for (int off = 16; off >= 1; off >>= 1)
#pragma unroll
        for (int e = 0; e < NEXP; ++e)
            acc[e] += __shfl_xor(acc[e], off, 32);

    float s[NEXP];
#pragma unroll
    for (int e = 0; e < NEXP; ++e) s[e] = acc[e] + br[e];

    float best1 = s[0]; int i1 = 0;
#pragma unroll
    for (int e = 1; e < NEXP; ++e)
        if (s[e] > best1) { best1 = s[e]; i1 = e; }
    float best2 = -3.4e38f; int i2 = -1;
#pragma unroll
    for (int e = 0; e < NEXP; ++e)
        if (e != i1 && s[e] > best2) { best2 = s[e]; i2 = e; }

    float m[NEXP];
#pragma unroll
    for (int e = 0; e < NEXP; ++e)
        m[e] = (e == i1) ? best1 : ((e == i2) ? best2 : 0.0f);
    float mx = m[0];
#pragma unroll
    for (int e = 1; e < NEXP; ++e) mx = fmaxf(mx, m[e]);
    float w[NEXP]; float sum = 0.f;
#pragma unroll
    for (int e = 0; e < NEXP; ++e) { w[e] = __expf(m[e] - mx); sum += w[e]; }
    const float inv = 1.0f / sum;
    if (lane == 0) {
#pragma unroll
        for (int e = 0; e < NEXP; ++e)
            wts[(size_t)t * NEXP + e] = w[e] * inv;
    }
}

// =====================================================================
// out[t,d] = sum_e w[t,e] * b2[e,d]   (bias part of weighted combine)
// =====================================================================
__global__ __launch_bounds__(256) void moe_init_out(
    const float* __restrict__ wts, const float* __restrict__ b2,
    float* __restrict__ out)
{
    const int idx = blockIdx.x * 256 + threadIdx.x;
    const int t = idx >> 10;
    const int d = idx & (EMBED - 1);
    float s = 0.f;
#pragma unroll
    for (int e = 0; e < NEXP; ++e)
        s = fmaf(wts[(size_t)t * NEXP + e], b2[(size_t)e * EMBED + d], s);
    out[idx] = s;
}

// =====================================================================
// Phase 1: H[t,j] = relu( x[t,:] @ W1[e][:, h0+j] + b1[e][h0+j] ) * w[t,e]
// =====================================================================
template<int HC>
__global__ __launch_bounds__(256) void moe_ffn1(
    const float* __restrict__ x, const float* __restrict__ W1,
    const float* __restrict__ b1, const float* __restrict__ wts,
    float* __restrict__ Hbuf, int e, int h0)
{
    __shared__ float As[2 * ABUF];
    __shared__ float Bs[2 * BBUF];
    __shared__ float wrow[128];

    const int tid  = threadIdx.x;
    const int lane = tid & 31;
    const int wave = tid >> 5;
    const int hh   = lane >> 4;
    const int l15  = lane & 15;
    const int n0   = blockIdx.x * 128;    // chunk-local hidden col base
    const int t0   = blockIdx.y * 128;    // token base
    const int jw   = wave * 16 + l15;

    if (tid < 128) wrow[tid] = wts[(size_t)(t0 + tid) * NEXP + e];

    const float* Ag = x + (size_t)t0 * EMBED;
    const float* Bg = W1 + (size_t)e * EMBED * HIDDEN + (size_t)(h0 + n0);

    v8f acc[8] = {};
    gemm_pipeline(Ag, EMBED, Bg, HIDDEN, EMBED, As, Bs, acc, tid, l15, hh, jw);

    const float bias = b1[(size_t)e * HIDDEN + h0 + n0 + jw];
    float* Hp = Hbuf + (size_t)t0 * HC + n0 + jw;
#pragma unroll
    for (int mt = 0; mt < 8; ++mt)
#pragma unroll
        for (int v = 0; v < 8; ++v) {
            const int row = mt * 16 + v + 8 * hh;   // C/D: lane>=16 -> M=v+8
            Hp[(size_t)row * HC] = fmaxf(acc[mt][v] + bias, 0.0f) * wrow[row];
        }
}

// =====================================================================
// Phase 2: out[t,d] += H[t,:] @ W2[e][h0:h0+HC, d]
// =====================================================================
template<int HC>
__global__ __launch_bounds__(256) void moe_ffn2(
    const float* __restrict__ Hbuf, const float* __restrict__ W2,
    float* __restrict__ out, int e, int h0)
{
    __shared__ float As[2 * ABUF];
    __shared__ float Bs[2 * BBUF];

    const int tid  = threadIdx.x;
    const int lane = tid & 31;
    const int wave = tid >> 5;
    const int hh   = lane >> 4;
    const int l15  = lane & 15;
    const int n0   = blockIdx.x * 128;    // embed col base
    const int t0   = blockIdx.y * 128;    // token base
    const int jw   = wave * 16 + l15;

    const float* Ag = Hbuf + (size_t)t0 * HC;
    const float* Bg = W2 + ((size_t)e * HIDDEN + (size_t)h0) * EMBED + n0;

    v8f acc[8] = {};
    gemm_pipeline(Ag, HC, Bg, EMBED, HC, As, Bs, acc, tid, l15, hh, jw);

    float* Op = out + (size_t)t0 * EMBED + n0 + jw;
#pragma unroll
    for (int mt = 0; mt < 8; ++mt)
#pragma unroll
        for (int v = 0; v < 8; ++v) {
            const int row = mt * 16 + v + 8 * hh;
            Op[(size_t)row * EMBED] += acc[mt][v];
        }
}

// =====================================================================
template<int HC>
static void run_experts(const float* x, const float* W1, const float* b1,
                        const float* W2, const float* wts, float* Hbuf,
                        float* out, hipStream_t stream)
{
    for (int e = 0; e < NEXP; ++e) {
        for (int h0 = 0; h0 < HIDDEN; h0 += HC) {
            moe_ffn1<HC><<<dim3(HC / 128, NTOK / 128), 256, 0, stream>>>(
                x, W1, b1, wts, Hbuf, e, h0);
            moe_ffn2<HC><<<dim3(EMBED / 128, NTOK / 128), 256, 0, stream>>>(
                Hbuf, W2, out, e, h0);
        }
    }
}

extern "C" void kernel_launch(void* const* d_in, const int* in_sizes, int n_in,
                              void* d_out, int out_size, void* d_ws, size_t ws_size,
                              hipStream_t stream)
{
    const float* x  = (const float*)d_in[0];
    const float* W1 = (const float*)d_in[1];
    const float* b1 = (const float*)d_in[2];
    const float* W2 = (const float*)d_in[3];
    const float* b2 = (const float*)d_in[4];
    const float* Wr = (const float*)d_in[5];
    const float* br = (const float*)d_in[6];
    float* out = (float*)d_out;

    // pick largest H-chunk that fits the workspace: H buffer + 128KB router weights
    const size_t wbytes = (size_t)NTOK * NEXP * sizeof(float);
    int hc = 512;
    if (ws_size == 0 || ws_size >= (size_t)NTOK * 2048 * sizeof(float) + wbytes) hc = 2048;
    else if (ws_size >= (size_t)NTOK * 1024 * sizeof(float) + wbytes)            hc = 1024;

    float* Hbuf = (float*)d_ws;
    float* wts  = (float*)((char*)d_ws + (size_t)NTOK * hc * sizeof(float));

    moe_router<<<NTOK / 8, 256, 0, stream>>>(x, Wr, br, wts);
    moe_init_out<<<(NTOK * EMBED) / 256, 256, 0, stream>>>(wts, b2, out);

    if (hc == 2048)      run_experts<2048>(x, W1, b1, W2, wts, Hbuf, out, stream);
    else if (hc == 1024) run_experts<1024>(x, W1, b1, W2, wts, Hbuf, out, stream);
    else                 run_experts<512 >(x, W1, b1, W2, wts, Hbuf, out, stream);
}